// MaskedBatchNorm2d_55490977464405
// MI455X (gfx1250) — compile-verified
//
#include <hip/hip_runtime.h>

// Problem constants from the reference: x[B=16, C=64, W=256, H=256] f32.
#define B_   16
#define C_   64
#define WH_  65536          // 256*256, power of two: b = s>>16, wh = s&0xFFFF
#define EPS_ 0.001f

typedef __attribute__((ext_vector_type(2))) float v2f;   // 16x4 f32 A/B operand (2 VGPRs)
typedef __attribute__((ext_vector_type(8))) float v8f;   // 16x16 f32 C/D operand (8 VGPRs)

// Full-wave (32 lane) sum using V_WMMA_F32_16X16X4_F32.
// Step 1: A = {val, 0} per lane -> A[m][0]=val[m], A[m][2]=val[m+16]; B = ones.
//         D[m][n] = val[m] + val[m+16].  D vgpr j: lanes0-15 = s[j], lanes16-31 = s[j+8].
// Step 2: feed D vgpr pairs back as A operands -> 4 wmmas give 4 partial totals
//         broadcast to all lanes; 3 adds finish. Only relies on the documented
//         A and C/D layouts (all-ones B makes the B layout irrelevant).
__device__ __forceinline__ float wave_reduce_f32_wmma(float v) {
  v2f a;    a[0] = v;    a[1] = 0.0f;
  v2f ones; ones[0] = 1.0f; ones[1] = 1.0f;
  v8f z = {};
  v8f d  = __builtin_amdgcn_wmma_f32_16x16x4_f32(false, a,  false, ones, (short)0, z, false, false);
  v2f p0; p0[0] = d[0]; p0[1] = d[1];
  v2f p1; p1[0] = d[2]; p1[1] = d[3];
  v2f p2; p2[0] = d[4]; p2[1] = d[5];
  v2f p3; p3[0] = d[6]; p3[1] = d[7];
  v8f t0 = __builtin_amdgcn_wmma_f32_16x16x4_f32(false, p0, false, ones, (short)0, z, false, false);
  v8f t1 = __builtin_amdgcn_wmma_f32_16x16x4_f32(false, p1, false, ones, (short)0, z, false, false);
  v8f t2 = __builtin_amdgcn_wmma_f32_16x16x4_f32(false, p2, false, ones, (short)0, z, false, false);
  v8f t3 = __builtin_amdgcn_wmma_f32_16x16x4_f32(false, p3, false, ones, (short)0, z, false, false);
  return (t0[0] + t1[0]) + (t2[0] + t3[0]);
}

// ---- Kernel 0: zero the accumulators (must run every launch: graph replays) ----
__global__ void mbn_init_k(float* __restrict__ S1, float* __restrict__ S2,
                           unsigned* __restrict__ nb) {
  int t = threadIdx.x;
  if (t < C_) { S1[t] = 0.0f; S2[t] = 0.0f; }
  if (t < B_) nb[t] = 0u;
}

// ---- Kernel 1: mask per spatial location + per-batch count ----
// Each thread handles 4 consecutive wh (float4): every one of the 64 channel
// loads is a 512B b128 wave transaction (channel stride = 256KB). Block covers
// 1024 locations; 65536 % 1024 == 0 so b is block-uniform.
__global__ void mbn_mask_k(const float* __restrict__ x, unsigned char* __restrict__ mf,
                           unsigned* __restrict__ nb) {
  unsigned q  = blockIdx.x * 256u + threadIdx.x;   // quad index over B*WH/4
  unsigned s  = q << 2;
  unsigned b  = s >> 16;
  unsigned wh = s & (WH_ - 1);
  const float* xb = x + (((size_t)b * C_) << 16) + wh;
  float4 acc = make_float4(0.f, 0.f, 0.f, 0.f);
#pragma unroll 8
  for (int c = 0; c < C_; ++c) {
    float4 v = *(const float4*)(xb + ((size_t)c << 16));
    acc.x += v.x; acc.y += v.y; acc.z += v.z; acc.w += v.w;
  }
  uchar4 mo;
  mo.x = (acc.x != 0.0f); mo.y = (acc.y != 0.0f);
  mo.z = (acc.z != 0.0f); mo.w = (acc.w != 0.0f);
  *(uchar4*)(mf + s) = mo;

  __shared__ unsigned cnt;
  if (threadIdx.x == 0) cnt = 0u;
  __syncthreads();
  atomicAdd(&cnt, (unsigned)mo.x + mo.y + mo.z + mo.w);   // LDS atomic
  __syncthreads();
  if (threadIdx.x == 0) atomicAdd(&nb[b], cnt);           // one global atomic per block
}

// ---- Kernel 2: per-channel S1 = sum(x*m), S2 = sum(x^2*m) ----
// grid = (spatial_chunks=256, C). Each block: channel c, 4096 contiguous
// locations in ONE batch (4096 | 65536), float4-vectorized, 16 elems/thread.
// Wave reduction via WMMA, then one atomic pair per block. mf (1MB) stays
// L2-resident across all 64 channel sweeps.
__global__ void mbn_stats_k(const float* __restrict__ x, const unsigned char* __restrict__ mf,
                            float* __restrict__ S1, float* __restrict__ S2) {
  const unsigned c     = blockIdx.y;
  const unsigned sBase = blockIdx.x * 4096u;       // block-uniform batch
  const unsigned b     = sBase >> 16;
  const unsigned whB   = sBase & (WH_ - 1);
  const float*  xp = x + (((size_t)(b * C_ + c)) << 16) + whB;
  const uchar4* mp = (const uchar4*)(mf + sBase);

  float a1 = 0.0f, a2 = 0.0f;
#pragma unroll
  for (unsigned i = threadIdx.x; i < 1024u; i += 256u) {   // uniform trips: EXEC stays full
    float4 v = *(const float4*)(xp + (i << 2));
    uchar4 m = mp[i];
    float w0 = (float)m.x, w1 = (float)m.y, w2 = (float)m.z, w3 = (float)m.w;
    float vw0 = v.x * w0, vw1 = v.y * w1, vw2 = v.z * w2, vw3 = v.w * w3;
    a1 += (vw0 + vw1) + (vw2 + vw3);
    a2 = fmaf(vw0, v.x, fmaf(vw1, v.y, fmaf(vw2, v.z, fmaf(vw3, v.w, a2))));
  }
  float r1v = wave_reduce_f32_wmma(a1);  // broadcast to all lanes
  float r2v = wave_reduce_f32_wmma(a2);

  __shared__ float r1[8], r2[8];
  unsigned wave = threadIdx.x >> 5;
  if ((threadIdx.x & 31u) == 0u) { r1[wave] = r1v; r2[wave] = r2v; }
  __syncthreads();
  if (threadIdx.x == 0) {
    float t1 = 0.0f, t2 = 0.0f;
#pragma unroll
    for (int i = 0; i < 8; ++i) { t1 += r1[i]; t2 += r2[i]; }
    atomicAdd(&S1[c], t1);
    atomicAdd(&S2[c], t2);
  }
}

// ---- Kernel 3: finalize mean/var/inv with padding correction (1 block, 64 threads) ----
// var numerator = S2 - 2*mean*S1 + mean^2*Nmask + sum_b pad[b]*(x00[b,c]-mean)^2
__global__ void mbn_finalize_k(const float* __restrict__ x, const float* __restrict__ S1,
                               const float* __restrict__ S2, const unsigned* __restrict__ nb,
                               float* __restrict__ inv, unsigned* __restrict__ hp) {
  int c = threadIdx.x;                 // 64 threads, one channel each
  unsigned Nmax = 0u;
#pragma unroll
  for (int b = 0; b < B_; ++b) Nmax = (nb[b] > Nmax) ? nb[b] : Nmax;

  float nmask = 0.0f;
  float pad[B_];
  float x00[B_];
#pragma unroll
  for (int b = 0; b < B_; ++b) {
    nmask += (float)nb[b];
    pad[b] = (float)(Nmax - nb[b]);
    x00[b] = x[((size_t)(b * C_ + c)) << 16];   // x[b, c, 0, 0]
  }
  float total = (float)(B_ * Nmax);
  float s1 = S1[c], s2 = S2[c];

  float sum_c = s1;
#pragma unroll
  for (int b = 0; b < B_; ++b) sum_c = fmaf(x00[b], pad[b], sum_c);
  float mean = sum_c / total;

  float varnum = s2 - 2.0f * mean * s1 + mean * mean * nmask;
#pragma unroll
  for (int b = 0; b < B_; ++b) {
    float d = x00[b] - mean;
    varnum = fmaf(pad[b] * d, d, varnum);
  }
  float var = varnum / total;
  inv[c] = 1.0f / sqrtf(var + EPS_);
  if (c < B_) hp[c] = (nb[c] < Nmax) ? 1u : 0u;   // has_pad per batch
}

// ---- Kernel 4: normalize (float4 vectorized; 16B coalesced loads/stores) ----
// out = m ? x*inv[c] : x;  at wh==0 also apply if has_pad[b] (x there == x00).
__global__ void mbn_norm_k(const float* __restrict__ x, const unsigned char* __restrict__ mf,
                           const float* __restrict__ inv, const unsigned* __restrict__ hp,
                           float* __restrict__ out) {
  unsigned e4 = blockIdx.x * 256u + threadIdx.x;
  size_t   e  = (size_t)e4 << 2;                  // element index, multiple of 4
  unsigned wh = (unsigned)(e & (WH_ - 1));
  unsigned cb = (unsigned)(e >> 16);
  unsigned c  = cb & (C_ - 1);
  unsigned b  = cb >> 6;

  float4 v = *(const float4*)(x + e);
  uchar4 m = *(const uchar4*)(mf + (((size_t)b) << 16) + wh);   // wh%4==0 -> aligned
  float iv = inv[c];
  bool pad0 = (wh == 0u) && (hp[b] != 0u);

  float4 o;
  o.x = (m.x || pad0) ? v.x * iv : v.x;
  o.y = m.y ? v.y * iv : v.y;
  o.z = m.z ? v.z * iv : v.z;
  o.w = m.w ? v.w * iv : v.w;
  *(float4*)(out + e) = o;
}

extern "C" void kernel_launch(void* const* d_in, const int* in_sizes, int n_in,
                              void* d_out, int out_size, void* d_ws, size_t ws_size,
                              hipStream_t stream) {
  const float* x   = (const float*)d_in[0];
  float*       out = (float*)d_out;

  // Workspace layout: 1MB mask + small accumulators (896B). 4B alignment holds.
  unsigned char* mf  = (unsigned char*)d_ws;
  float*         S1  = (float*)(mf + (size_t)B_ * WH_);
  float*         S2  = S1 + C_;
  float*         inv = S2 + C_;
  unsigned*      nb  = (unsigned*)(inv + C_);
  unsigned*      hp  = nb + B_;

  mbn_init_k<<<1, 256, 0, stream>>>(S1, S2, nb);
  mbn_mask_k<<<(B_ * WH_) / (4 * 256), 256, 0, stream>>>(x, mf, nb);
  mbn_stats_k<<<dim3((B_ * WH_) / 4096, C_), 256, 0, stream>>>(x, mf, S1, S2);
  mbn_finalize_k<<<1, C_, 0, stream>>>(x, S1, S2, nb, inv, hp);
  mbn_norm_k<<<(B_ * C_ * WH_) / (4 * 256), 256, 0, stream>>>(x, mf, inv, hp, out);
}